// STM_87273735455328
// MI455X (gfx1250) — compile-verified
//
#include <hip/hip_runtime.h>
#include <cstdint>
#include <cstddef>

// Problem constants (fixed by the reference)
#define B_ROWS 4096
#define D_DIM  2048
#define K_DIM  16384
// SIDE = 128 -> gx = k >> 7, gy = k & 127

// GEMM tiling: 128x128 C tile per 256-thread block, 8 wave32s in 4x2,
// each wave owns 32 rows x 64 cols = eight 16x16 f32 accumulators.
#define BM 128
#define BN 128
#define KB 32
#define AS_STRIDE 36    // 128 x 32 A tile, padded rows (36*4 = 144 B, 16B aligned)
#define BP_STRIDE 264   // per k-pair row: 128 cols * 2 interleaved + 8 pad floats

typedef float v2f __attribute__((ext_vector_type(2)));
typedef float v8f __attribute__((ext_vector_type(8)));

// ---------------------------------------------------------------------------
// fp32 WMMA GEMM over one 128x128 C tile: acc = X[rows,:] @ W[:,cols]
// A tile row-major (k contiguous -> b64 fragment loads); B tile stored
// k-pair interleaved so each 4x16 fragment is ONE contiguous ds_load_b64.
// ---------------------------------------------------------------------------
__device__ __forceinline__ void gemm_tile_f32(
    const float* __restrict__ X, const float* __restrict__ W,
    int rowBase, int colBase, float* As, float* Bs, v8f acc[8])
{
    const int tid   = threadIdx.x;
    const int lane  = tid & 31;
    const int wave  = tid >> 5;
    const int waveM = wave & 3;     // row strip 0..3 (32 rows each)
    const int waveC = wave >> 2;    // col strip 0..1 (64 cols each)
    const int m     = lane & 15;    // lane within half-wave
    const int h     = lane >> 4;    // half: selects K-pair {0,1} vs {2,3}
    const int rTile = waveM * 32;
    const int cTile = waveC * 64;

    #pragma unroll
    for (int i = 0; i < 8; ++i) {
        v8f z = {0.f, 0.f, 0.f, 0.f, 0.f, 0.f, 0.f, 0.f};
        acc[i] = z;
    }

    for (int kt = 0; kt < D_DIM; kt += KB) {
        __syncthreads();  // protect previous iteration's LDS reads

        // ---- stage A: 128x32 floats = 1024 float4, 4 per thread ----
        #pragma unroll
        for (int i = 0; i < 4; ++i) {
            int f4 = tid + 256 * i;         // 0..1023
            int r  = f4 >> 3;               // row 0..127
            int c4 = f4 & 7;                // float4 col 0..7
            float4 v = *(const float4*)&X[(size_t)(rowBase + r) * D_DIM + kt + c4 * 4];
            *(float4*)&As[r * AS_STRIDE + c4 * 4] = v;
        }
        // ---- stage B: 32x128 floats, k-pair interleaved: Bs[k/2][n*2 + (k&1)] ----
        #pragma unroll
        for (int i = 0; i < 4; ++i) {
            int f4 = tid + 256 * i;         // 0..1023
            int r  = f4 >> 5;               // k row 0..31
            int c4 = f4 & 31;               // float4 col 0..31
            const float* gp = &W[(size_t)(kt + r) * K_DIM + colBase + c4 * 4];
            float4 v = *(const float4*)gp;
            float* bp = &Bs[(r >> 1) * BP_STRIDE + (r & 1)];
            bp[(c4 * 4 + 0) * 2] = v.x;
            bp[(c4 * 4 + 1) * 2] = v.y;
            bp[(c4 * 4 + 2) * 2] = v.z;
            bp[(c4 * 4 + 3) * 2] = v.w;
            if (kt + KB < D_DIM) {
                // speculative prefetch of next k-tile -> global_prefetch_b8
                __builtin_prefetch(gp + (size_t)KB * K_DIM, 0, 0);
            }
        }
        __syncthreads();

        // ---- compute: 8 k-steps x 8 WMMAs per wave ----
        #pragma unroll
        for (int kk = 0; kk < KB; kk += 4) {
            // A frags 16x4: lane m holds row M=m; h selects K-pair {kk,kk+1}/{kk+2,kk+3}
            v2f a0, a1;
            a0[0] = As[(rTile + m) * AS_STRIDE + kk + 2 * h + 0];
            a0[1] = As[(rTile + m) * AS_STRIDE + kk + 2 * h + 1];
            a1[0] = As[(rTile + 16 + m) * AS_STRIDE + kk + 2 * h + 0];
            a1[1] = As[(rTile + 16 + m) * AS_STRIDE + kk + 2 * h + 1];
            const int p = (kk >> 1) + h;    // k-pair index for this half-wave
            #pragma unroll
            for (int f = 0; f < 4; ++f) {
                const float* bp = &Bs[p * BP_STRIDE + (cTile + f * 16 + m) * 2];
                v2f b;
                b[0] = bp[0];               // contiguous pair -> single ds_load_b64
                b[1] = bp[1];
                acc[f] = __builtin_amdgcn_wmma_f32_16x16x4_f32(
                    false, a0, false, b, (short)0, acc[f], false, false);
                acc[4 + f] = __builtin_amdgcn_wmma_f32_16x16x4_f32(
                    false, a1, false, b, (short)0, acc[4 + f], false, false);
            }
        }
    }
}

// ---------------------------------------------------------------------------
// Pass 1: GEMM + per-row argmin of norms2 via packed u64 atomic min.
// key = (float_bits(norms2) << 32) | col  (norms2 >= 0 -> bits order-preserving;
// index in low bits reproduces argmin's first-hit tie-break)
// ---------------------------------------------------------------------------
__global__ __launch_bounds__(256) void stm_argmin_kernel(
    const float* __restrict__ X, const float* __restrict__ W,
    const float* __restrict__ x2, const float* __restrict__ k2,
    unsigned long long* __restrict__ rowmin)
{
    __shared__ float As[BM * AS_STRIDE];
    __shared__ float Bs[(KB / 2) * BP_STRIDE];
    __shared__ float x2s[BM];
    __shared__ float k2s[BN];

    const int rowBase = blockIdx.y * BM;
    const int colBase = blockIdx.x * BN;
    const int tid = threadIdx.x;
    if (tid < BM) x2s[tid] = x2[rowBase + tid];
    if (tid < BN) k2s[tid] = k2[colBase + tid];

    v8f acc[8];
    gemm_tile_f32(X, W, rowBase, colBase, As, Bs, acc);

    const int lane  = tid & 31;
    const int wave  = tid >> 5;
    const int waveM = wave & 3;
    const int waveC = wave >> 2;
    const int m     = lane & 15;
    const int h     = lane >> 4;
    const int rTile = waveM * 32;
    const int cTile = waveC * 64;

    #pragma unroll
    for (int rr = 0; rr < 2; ++rr) {
        #pragma unroll
        for (int r = 0; r < 8; ++r) {
            const int rowL = rTile + rr * 16 + h * 8 + r;  // C layout: half 1 -> M=8+r
            const float xx = x2s[rowL];
            float vmin = 3.402823466e38f;
            int   cmin = 0x7FFFFFFF;
            #pragma unroll
            for (int f = 0; f < 4; ++f) {
                const int colL = cTile + f * 16 + m;
                const float n2 = fmaxf(xx - 2.0f * acc[rr * 4 + f][r] + k2s[colL], 0.0f);
                if (n2 < vmin) { vmin = n2; cmin = colBase + colL; }  // first index wins
            }
            // reduce across the 16 lanes of this half-wave (all hold the same row)
            #pragma unroll
            for (int s = 8; s > 0; s >>= 1) {
                float ov = __shfl_xor(vmin, s, 16);
                int   oc = __shfl_xor(cmin, s, 16);
                if (ov < vmin || (ov == vmin && oc < cmin)) { vmin = ov; cmin = oc; }
            }
            if (m == 0) {
                unsigned long long key =
                    ((unsigned long long)__float_as_uint(vmin) << 32) | (unsigned int)cmin;
                atomicMin(&rowmin[rowBase + rowL], key);
            }
        }
    }
}

// ---------------------------------------------------------------------------
// Pass 2: recompute GEMM (x+kernel = 167MB fits the 192MB L2 -> cheap re-read),
// fuse the radial epilogue into the single 268MB output write.
// ---------------------------------------------------------------------------
__global__ __launch_bounds__(256) void stm_output_kernel(
    const float* __restrict__ X, const float* __restrict__ W,
    const float* __restrict__ x2, const float* __restrict__ k2,
    const unsigned long long* __restrict__ rowmin,
    const float* __restrict__ sigma_p,
    float* __restrict__ out)
{
    __shared__ float As[BM * AS_STRIDE];
    __shared__ float Bs[(KB / 2) * BP_STRIDE];
    __shared__ float x2s[BM];
    __shared__ float k2s[BN];
    __shared__ float wxs[BM];
    __shared__ float wys[BM];

    const int rowBase = blockIdx.y * BM;
    const int colBase = blockIdx.x * BN;
    const int tid = threadIdx.x;
    if (tid < BM) {
        x2s[tid] = x2[rowBase + tid];
        unsigned int idx = (unsigned int)(rowmin[rowBase + tid] & 0xFFFFFFFFu);
        wxs[tid] = (float)(idx >> 7);     // idx / SIDE
        wys[tid] = (float)(idx & 127);    // idx % SIDE
    }
    if (tid < BN) k2s[tid] = k2[colBase + tid];

    const float sigma  = *sigma_p;
    const float inv2s2 = -0.5f / (sigma * sigma);
    const float scale  = 1.0f / (sigma * 2.5066282746310002f);  // sigma*sqrt(2*pi)

    v8f acc[8];
    gemm_tile_f32(X, W, rowBase, colBase, As, Bs, acc);

    const int lane  = tid & 31;
    const int wave  = tid >> 5;
    const int waveM = wave & 3;
    const int waveC = wave >> 2;
    const int m     = lane & 15;
    const int h     = lane >> 4;
    const int rTile = waveM * 32;
    const int cTile = waveC * 64;

    #pragma unroll
    for (int rr = 0; rr < 2; ++rr) {
        #pragma unroll
        for (int r = 0; r < 8; ++r) {
            const int rowL = rTile + rr * 16 + h * 8 + r;
            const int rowG = rowBase + rowL;
            const float xx = x2s[rowL];
            const float wx = wxs[rowL];
            const float wy = wys[rowL];
            #pragma unroll
            for (int f = 0; f < 4; ++f) {
                const int colL = cTile + f * 16 + m;
                const int colG = colBase + colL;
                const float n2 = fmaxf(xx - 2.0f * acc[rr * 4 + f][r] + k2s[colL], 0.0f);
                const float gx = (float)(colG >> 7);
                const float gy = (float)(colG & 127);
                const float dx = gx - wx;
                const float dy = gy - wy;
                const float rad = expf(inv2s2 * (dx * dx + dy * dy)) * scale;
                out[(size_t)rowG * K_DIM + colG] = rad * n2;
            }
        }
    }
}

// ---------------------------------------------------------------------------
// Helpers: init argmin keys, row norms of x, column norms of kernel
// ---------------------------------------------------------------------------
__global__ void init_rowmin_kernel(unsigned long long* __restrict__ rowmin) {
    int i = blockIdx.x * blockDim.x + threadIdx.x;
    if (i < B_ROWS) rowmin[i] = ~0ull;
}

__global__ __launch_bounds__(256) void row_sq_kernel(
    const float* __restrict__ X, float* __restrict__ x2)
{
    const int wave = (blockIdx.x * blockDim.x + threadIdx.x) >> 5;
    const int lane = threadIdx.x & 31;
    if (wave >= B_ROWS) return;
    const float* row = X + (size_t)wave * D_DIM;
    float s = 0.f;
    for (int d = lane; d < D_DIM; d += 32) { float v = row[d]; s = fmaf(v, v, s); }
    #pragma unroll
    for (int msk = 16; msk; msk >>= 1) s += __shfl_xor(s, msk, 32);
    if (lane == 0) x2[wave] = s;
}

__global__ __launch_bounds__(256) void col_sq_kernel(
    const float* __restrict__ W, float* __restrict__ k2)
{
    const int k = blockIdx.x * blockDim.x + threadIdx.x;
    if (k >= K_DIM) return;
    float s = 0.f;
    for (int d = 0; d < D_DIM; ++d) { float v = W[(size_t)d * K_DIM + k]; s = fmaf(v, v, s); }
    k2[k] = s;
}

// ---------------------------------------------------------------------------
extern "C" void kernel_launch(void* const* d_in, const int* in_sizes, int n_in,
                              void* d_out, int out_size, void* d_ws, size_t ws_size,
                              hipStream_t stream) {
    const float* X   = (const float*)d_in[0];   // [4096, 2048]
    const float* W   = (const float*)d_in[1];   // [2048, 16384]
    const float* sig = (const float*)d_in[2];   // scalar
    float* out = (float*)d_out;                 // [4096, 16384]

    // workspace layout: rowmin (32 KB) | x2 (16 KB) | k2 (64 KB) = 112 KB
    char* ws = (char*)d_ws;
    unsigned long long* rowmin = (unsigned long long*)ws;
    float* x2 = (float*)(ws + 32768);
    float* k2 = (float*)(ws + 49152);

    init_rowmin_kernel<<<(B_ROWS + 255) / 256, 256, 0, stream>>>(rowmin);
    row_sq_kernel<<<B_ROWS / 8, 256, 0, stream>>>(X, x2);        // one wave32 per row
    col_sq_kernel<<<K_DIM / 256, 256, 0, stream>>>(W, k2);

    dim3 grid(K_DIM / BN, B_ROWS / BM);   // 128 x 32 = 4096 blocks
    stm_argmin_kernel<<<grid, 256, 0, stream>>>(X, W, x2, k2, rowmin);
    stm_output_kernel<<<grid, 256, 0, stream>>>(X, W, x2, k2, rowmin, sig, out);
}